// ProgressReward_44787918963377
// MI455X (gfx1250) — compile-verified
//
#include <hip/hip_runtime.h>
#include <stdint.h>

#define NUM_HIST_C 4
#define INTERVAL_C 5
#define TILE 256
#define MAXTHREADS 256
#define MAXQ 64

// ---- CDNA5 async global->LDS copy (ASYNCcnt path), inline asm ----
__device__ __forceinline__ void async_ld_f4(unsigned lds_off, unsigned long long gaddr) {
    asm volatile("global_load_async_to_lds_b128 %0, %1, off"
                 :: "v"(lds_off), "v"(gaddr) : "memory");
}
__device__ __forceinline__ void wait_async0() {
    asm volatile("s_wait_asynccnt 0" ::: "memory");
}

// ---- Kernel 1: pack polyline state {cx, cy, cos, sin}; off-route -> sentinel ----
__global__ void pr_prep_kernel(const float* __restrict__ pos,
                               const float* __restrict__ head,
                               const int*   __restrict__ edge_row1,
                               const int*   __restrict__ on_route,
                               float4* __restrict__ out, int P) {
    int p = blockIdx.x * blockDim.x + threadIdx.x;
    if (p >= P) return;
    int e = edge_row1[p];
    float4 v;
    if (on_route[e] != 0) {
        float h = head[p];
        v.x = pos[2 * p + 0];
        v.y = pos[2 * p + 1];
        v.z = cosf(h);
        v.w = sinf(h);
    } else {
        // rotation preserves norm -> dist >= ~7e29, never beats any valid poly
        v.x = 1e30f; v.y = 0.0f; v.z = 1.0f; v.w = 0.0f;
    }
    out[p] = v;
}

// ---- Kernel 2: per-batch ranges via binary search on sorted polyline_batch ----
__global__ void pr_range_kernel(const int* __restrict__ pb, int P, int B,
                                int* __restrict__ starts) {
    int b = threadIdx.x;
    if (b > B) return;
    int lo = 0, hi = P;
    while (lo < hi) {            // lower_bound(b)
        int m = (lo + hi) >> 1;
        if (pb[m] < b) lo = m + 1; else hi = m;
    }
    starts[b] = lo;
}

// ---- Kernel 3: main progress/argmin kernel, one block per batch ----
__global__ void pr_main_kernel(const float4* __restrict__ poly,
                               const int*   __restrict__ starts,
                               const int*   __restrict__ agent_ptr,
                               const int*   __restrict__ agent_batch,
                               const float* __restrict__ infer_pos,
                               const float* __restrict__ agent_pos,
                               float* __restrict__ out,
                               int T, int steps, int L) {
    __shared__ float4 tile[2][TILE];
    __shared__ float  pd [MAXTHREADS];
    __shared__ float  px [MAXTHREADS];
    __shared__ float  pxp[MAXTHREADS];
    __shared__ int    pidx[MAXTHREADS];
    __shared__ float  qprog[MAXQ];

    const int b   = blockIdx.x;
    const int tid = threadIdx.x;
    const int q   = tid >> 2;      // query index: [0,T) infer, [T,2T) expert
    const int r4  = tid & 3;       // 4 scanning threads per query
    const int nq  = 2 * T;

    const int ego = agent_ptr[b];
    const int bb  = agent_batch[ego];   // == b in this config, but be faithful

    // Load this query's (pos, pos_pre)
    float qx = 0.f, qy = 0.f, qpx = 0.f, qpy = 0.f;
    if (q < nq) {
        int t = (q < T) ? q : (q - T);
        if (q < T) {
            const float* base = infer_pos + (size_t)ego * steps * 2;
            qx  = base[(NUM_HIST_C     + t) * 2 + 0];
            qy  = base[(NUM_HIST_C     + t) * 2 + 1];
            qpx = base[(NUM_HIST_C - 1 + t) * 2 + 0];
            qpy = base[(NUM_HIST_C - 1 + t) * 2 + 1];
        } else {
            // expert = agent_position[ego, ::5]; sample s -> row s*5
            const float* base = agent_pos + (size_t)ego * L * 2;
            qx  = base[((NUM_HIST_C     + t) * INTERVAL_C) * 2 + 0];
            qy  = base[((NUM_HIST_C     + t) * INTERVAL_C) * 2 + 1];
            qpx = base[((NUM_HIST_C - 1 + t) * INTERVAL_C) * 2 + 0];
            qpy = base[((NUM_HIST_C - 1 + t) * INTERVAL_C) * 2 + 1];
        }
    }

    const int s0 = starts[bb];
    const int s1 = starts[bb + 1];
    const int n  = s1 - s0;
    const int ntile = (n + TILE - 1) / TILE;

    float bd = 1e28f, bx = 0.f, bxp = 0.f;
    int   bi = 0x7fffffff;

    // Prefetch tile 0 (double-buffered async global->LDS)
    if (ntile > 0) {
        int cnt0 = min(TILE, n);
        if (tid < cnt0) {
            unsigned lo = (unsigned)(uintptr_t)&tile[0][tid];
            async_ld_f4(lo, (unsigned long long)(uintptr_t)(poly + s0 + tid));
        }
    }

    for (int k = 0; k < ntile; ++k) {
        wait_async0();        // my tile-k element landed in LDS
        __syncthreads();      // everyone's landed; previous buffer fully consumed

        const int tb  = s0 + k * TILE;
        const int cnt = min(TILE, s1 - tb);

        if (k + 1 < ntile) {  // kick tile k+1 into the other buffer
            int tb2  = tb + TILE;
            int cnt2 = min(TILE, s1 - tb2);
            if (tid < cnt2) {
                unsigned lo = (unsigned)(uintptr_t)&tile[(k + 1) & 1][tid];
                async_ld_f4(lo, (unsigned long long)(uintptr_t)(poly + tb2 + tid));
            }
        }

        const float4* buf = tile[k & 1];
        for (int j = r4; j < cnt; j += 4) {
            float4 pl = buf[j];
            float dx = qx - pl.x, dy = qy - pl.y;
            float x  =  pl.z * dx + pl.w * dy;
            float y  = -pl.w * dx + pl.z * dy;
            float d  = fabsf(y) * 10.0f + fabsf(x) + (x > 0.0f ? 1000.0f : 0.0f);
            if (d < bd) {     // strict < keeps first min within this thread
                bd = d; bi = tb + j; bx = x;
                float dxp = qpx - pl.x, dyp = qpy - pl.y;
                bxp = pl.z * dxp + pl.w * dyp;
            }
        }
    }

    pd[tid] = bd; px[tid] = bx; pxp[tid] = bxp; pidx[tid] = bi;
    __syncthreads();

    if (r4 == 0 && q < nq) {
        float d0 = pd[tid], X = px[tid], XP = pxp[tid];
        int   I  = pidx[tid];
        for (int r = 1; r < 4; ++r) {   // combine with first-index tie-break
            float d = pd[tid + r];
            int   i = pidx[tid + r];
            if (d < d0 || (d == d0 && i < I)) {
                d0 = d; X = px[tid + r]; XP = pxp[tid + r]; I = i;
            }
        }
        qprog[q] = (d0 < 1e28f) ? (X - XP) : 0.0f;  // has-valid gate
    }
    __syncthreads();

    if (tid == 0) {
        float pi = 0.f, pe = 0.f;
        for (int t = 0; t < T; ++t) pi += qprog[t];
        for (int t = 0; t < T; ++t) pe += qprog[T + t];
        out[b] = fminf(fmaxf(pi, 2.0f) / fmaxf(pe, 2.0f), 1.0f);
    }
}

extern "C" void kernel_launch(void* const* d_in, const int* in_sizes, int n_in,
                              void* d_out, int out_size, void* d_ws, size_t ws_size,
                              hipStream_t stream) {
    const int*   polyline_batch    = (const int*)  d_in[0];
    const float* polyline_position = (const float*)d_in[1];
    const float* polyline_heading  = (const float*)d_in[2];
    const int*   edge_index        = (const int*)  d_in[3];   // (2,P) row-major
    const int*   on_route_mask     = (const int*)  d_in[4];   // bool -> int per harness doc
    const int*   agent_ptr         = (const int*)  d_in[5];
    const float* infer_pos         = (const float*)d_in[6];   // (A, steps, 2)
    const int*   agent_batch       = (const int*)  d_in[7];
    const float* agent_pos         = (const float*)d_in[8];   // (A, L, 2)

    const int P     = in_sizes[0];
    const int B     = in_sizes[5] - 1;
    const int A     = in_sizes[7];
    const int steps = in_sizes[6] / (2 * A);      // = NUM_HIST + T = 36
    const int T     = steps - NUM_HIST_C;         // = 32
    const int L     = in_sizes[8] / (2 * A);      // = 180

    float4* poly_packed = (float4*)d_ws;
    int*    starts      = (int*)((char*)d_ws + (size_t)P * sizeof(float4));

    pr_prep_kernel<<<(P + 255) / 256, 256, 0, stream>>>(
        polyline_position, polyline_heading, edge_index + P, on_route_mask,
        poly_packed, P);

    pr_range_kernel<<<1, B + 1, 0, stream>>>(polyline_batch, P, B, starts);

    pr_main_kernel<<<B, 8 * T, 0, stream>>>(
        poly_packed, starts, agent_ptr, agent_batch, infer_pos, agent_pos,
        (float*)d_out, T, steps, L);
}